// Model_9225589752151
// MI455X (gfx1250) — compile-verified
//
#include <hip/hip_runtime.h>
#include <math.h>

// Parallel-beam Radon transform for MI455X (gfx1250).
//
// Strategy (see analysis): gather-bound problem -> stage the whole 256x256 f32
// image (257-float padded rows, 257KB) in the 320KB LDS of one WGP using
// CDNA5 async global->LDS loads, then each thread integrates one ray with
// branchless bilinear sampling from LDS. WMMA intentionally unused: no dense
// shared contraction exists; densifying the 2-banded interpolation operator
// would inflate FLOPs ~500x.

#define NPIX 256            // image side == detectors == ray steps
#define NANG 256            // projection angles
#define NB   8              // batch
#define ROWP 257            // padded LDS row stride (floats) -> bank-conflict free
#define ANG_PER_BLOCK 4
#define THREADS 1024        // 32 wave32 waves per workgroup

__global__ __launch_bounds__(THREADS)
void radon_fwd_kernel(const float* __restrict__ img,     // [B,1,N,N]
                      const float* __restrict__ angles,  // [A]
                      float* __restrict__ sino)          // [B,1,A,N]
{
    extern __shared__ float smem[];          // NPIX * ROWP floats (257KB)

    const int tid = threadIdx.x;
    const int b   = blockIdx.x / (NANG / ANG_PER_BLOCK);
    const int ag  = blockIdx.x % (NANG / ANG_PER_BLOCK);

    const float* gimg = img + (size_t)b * NPIX * NPIX;

    // ---- Async stage: global -> LDS, 1024 threads x 16 x b128 = 256KB ----
    // LDS byte offset of dynamic shared base (low 32 bits of flat address).
    const unsigned lds_base = (unsigned)(size_t)smem;
    #pragma unroll
    for (int i = 0; i < (NPIX * NPIX / 4) / THREADS; ++i) {
        int f    = tid + i * THREADS;        // float4 index, 64 float4 per row
        int row  = f >> 6;
        int col4 = f & 63;
        unsigned lds_byte = lds_base + (unsigned)(row * ROWP + col4 * 4) * 4u;
        int      goff     = f * 16;          // global byte offset (GVS vaddr)
        asm volatile(
            "global_load_async_to_lds_b128 %0, %1, %2"
            :
            : "v"(lds_byte), "v"(goff), "s"(gimg)
            : "memory");
    }
    asm volatile("s_wait_asynccnt 0" ::: "memory");
    __syncthreads();

    // ---- Ray integration: thread -> (local angle, detector) ----
    const int la  = tid >> 8;                // 0..3
    const int det = tid & 255;
    const int a   = ag * ANG_PER_BLOCK + la;

    const float th = angles[a];
    float sn, cs;
    sincosf(th, &sn, &cs);

    const float c  = (float)(NPIX - 1) * 0.5f;   // 127.5
    const float t  = (float)det - c;
    // X(s) = t*cos - s*sin + c ; Y(s) = t*sin + s*cos + c
    const float xb = fmaf(t, cs, c);
    const float yb = fmaf(t, sn, c);

    float acc = 0.0f;
    float sk  = -c;                          // s = k - 127.5
    #pragma unroll 4
    for (int k = 0; k < NPIX; ++k) {
        float x = fmaf(-sk, sn, xb);
        float y = fmaf( sk, cs, yb);

        float xf = floorf(x), yf = floorf(y);
        float fx = x - xf,    fy = y - yf;
        int x0 = (int)xf, y0 = (int)yf;
        int x1 = x0 + 1,  y1 = y0 + 1;

        // map_coordinates(order=1, mode=constant, cval=0): each OOB neighbor
        // contributes zero -> clamp index for the load, zero via mask weight.
        float mx0 = ((unsigned)x0 < (unsigned)NPIX) ? 1.0f : 0.0f;
        float mx1 = ((unsigned)x1 < (unsigned)NPIX) ? 1.0f : 0.0f;
        float my0 = ((unsigned)y0 < (unsigned)NPIX) ? 1.0f : 0.0f;
        float my1 = ((unsigned)y1 < (unsigned)NPIX) ? 1.0f : 0.0f;

        int cx0 = min(max(x0, 0), NPIX - 1);
        int cx1 = min(max(x1, 0), NPIX - 1);
        int ry0 = min(max(y0, 0), NPIX - 1) * ROWP;
        int ry1 = min(max(y1, 0), NPIX - 1) * ROWP;

        float v00 = smem[ry0 + cx0] * (my0 * mx0);
        float v01 = smem[ry0 + cx1] * (my0 * mx1);
        float v10 = smem[ry1 + cx0] * (my1 * mx0);
        float v11 = smem[ry1 + cx1] * (my1 * mx1);

        float top = fmaf(fx, v01 - v00, v00);
        float bot = fmaf(fx, v11 - v10, v10);
        acc = fmaf(fy, bot - top, top) + acc;

        sk += 1.0f;
    }

    sino[((size_t)b * NANG + a) * NPIX + det] = acc;
}

extern "C" void kernel_launch(void* const* d_in, const int* in_sizes, int n_in,
                              void* d_out, int out_size, void* d_ws, size_t ws_size,
                              hipStream_t stream) {
    (void)in_sizes; (void)n_in; (void)d_ws; (void)ws_size; (void)out_size;

    const float* x      = (const float*)d_in[0];   // [8,1,256,256] f32
    const float* angles = (const float*)d_in[1];   // [256] f32
    float*       out    = (float*)d_out;           // [8,1,256,256] f32

    const size_t shmem = (size_t)NPIX * ROWP * sizeof(float);  // 263168 B
    // Allow >64KB dynamic LDS (gfx1250 WGP has 320KB).
    (void)hipFuncSetAttribute((const void*)radon_fwd_kernel,
                              hipFuncAttributeMaxDynamicSharedMemorySize,
                              (int)shmem);

    dim3 grid(NB * (NANG / ANG_PER_BLOCK));        // 512 workgroups
    radon_fwd_kernel<<<grid, THREADS, shmem, stream>>>(x, angles, out);
}